// GarNetLayer_76690936037518
// MI455X (gfx1250) — compile-verified
//
#include <hip/hip_runtime.h>

typedef __attribute__((ext_vector_type(2))) float v2f;
typedef __attribute__((ext_vector_type(8))) float v8f;

namespace {
constexpr int V = 128;  // vertices (K dim)
constexpr int S = 16;   // aggregators (M dim)
constexpr int N = 64;   // features   (N dim)
}

// One workgroup per batch element. 128 threads = 4 wave32; wave w owns
// N-tile [16w, 16w+16). D = A(16xV) x B(VxN-tile) via V_WMMA_F32_16X16X4_F32.
__global__ __launch_bounds__(128, 1) void garnet_wmma_kernel(
    const float* __restrict__ fi_v,   // [B, V, N]
    const float* __restrict__ d_av,   // [B, V, S]
    float* __restrict__ out)          // [B, S*N]
{
  __shared__ float wlds[V * S];   // w[v][s] = exp(-d^2), row-major [V][S]
  __shared__ float psum[8 * S];   // partial column sums
  __shared__ float wscale[S];     // (sum_v w[v][s]) / V^2

  const int tid = threadIdx.x;    // 0..127
  const int b   = blockIdx.x;

  // ---- Phase 1: w = exp(-d*d); thread t computes full row v = t.
  // Coalesced: 64B contiguous per thread via 4x float4.
  {
    const float4* dp = reinterpret_cast<const float4*>(
        d_av + (size_t)b * (V * S) + (size_t)tid * S);
    float4* wp = reinterpret_cast<float4*>(&wlds[tid * S]);
#pragma unroll
    for (int i = 0; i < 4; ++i) {
      float4 d = dp[i];
      float4 w;
      w.x = __expf(-d.x * d.x);
      w.y = __expf(-d.y * d.y);
      w.z = __expf(-d.z * d.z);
      w.w = __expf(-d.w * d.w);
      wp[i] = w;
    }
  }
  __syncthreads();

  // ---- Phase 2: column sums of w -> wscale[s] = sum_v w[v][s] / (V*V)
  // (folds the 1/V of the mean-matmul and the 1/V of wbar together)
  {
    const int s = tid & 15;
    const int g = tid >> 4;       // 8 groups of 16 rows
    float acc = 0.f;
#pragma unroll
    for (int vv = 0; vv < 16; ++vv)
      acc += wlds[(g * 16 + vv) * S + s];
    psum[g * S + s] = acc;
  }
  __syncthreads();
  if (tid < S) {
    float acc = 0.f;
#pragma unroll
    for (int g = 0; g < 8; ++g) acc += psum[g * S + tid];
    wscale[tid] = acc * (1.0f / ((float)V * (float)V));
  }
  __syncthreads();

  // ---- Phase 3: C[s,n] = sum_v w[v][s] * fi[v][n] on the WMMA pipe.
  // ISA 32-bit fragment layouts (16x16x4):
  //   A (16x4):  lanes 0-15: M=lane, VGPR0=K+0, VGPR1=K+1
  //              lanes16-31: M=lane-16, VGPR0=K+2, VGPR1=K+3
  //   B (4x16):  lanes 0-15: N=lane,  VGPR0=K+0, VGPR1=K+1
  //              lanes16-31: N=lane-16, VGPR0=K+2, VGPR1=K+3
  const int lane = tid & 31;
  const int wv   = tid >> 5;          // wave id = N tile
  const int half = lane >> 4;         // 0 or 1
  const int m    = lane & 15;
  const int col  = wv * 16 + m;       // global N column
  const float* fib = fi_v + (size_t)b * (V * N) + col;

  v8f acc = {};
#pragma unroll
  for (int k = 0; k < V; k += 4) {
    const int r0 = k + 2 * half;
    v2f a, bm;
    a.x  = wlds[r0 * S + m];          // A[M=m, K=r0]   = w[r0][m]
    a.y  = wlds[(r0 + 1) * S + m];    // A[M=m, K=r0+1] = w[r0+1][m]
    bm.x = fib[r0 * N];               // B[K=r0,   N=col] = fi[r0][col]
    bm.y = fib[(r0 + 1) * N];         // B[K=r0+1, N=col]
    acc = __builtin_amdgcn_wmma_f32_16x16x4_f32(
        /*neg_a=*/false, a, /*neg_b=*/false, bm,
        /*c_mod=*/(short)0, acc, /*reuse_a=*/false, /*reuse_b=*/false);
  }

  // ---- Phase 4: out[s,n] = wscale[s] * C[s,n]
  // C/D layout: VGPR r -> row s = r + 8*half, col n = (lane&15) in tile.
  float* op = out + (size_t)b * (S * N) + (size_t)(8 * half) * N + col;
#pragma unroll
  for (int r = 0; r < 8; ++r) {
    op[r * N] = acc[r] * wscale[8 * half + r];
  }
}

extern "C" void kernel_launch(void* const* d_in, const int* in_sizes, int n_in,
                              void* d_out, int out_size, void* d_ws, size_t ws_size,
                              hipStream_t stream) {
  const float* fi_v = (const float*)d_in[0];  // [B,V,N] f32
  const float* d_av = (const float*)d_in[1];  // [B,V,S] f32
  float* out = (float*)d_out;                 // [B,S*N] f32
  const int B = in_sizes[0] / (V * N);        // 4096
  garnet_wmma_kernel<<<B, 128, 0, stream>>>(fi_v, d_av, out);
}